// QuantHotLowRank_87359634800682
// MI455X (gfx1250) — compile-verified
//
#include <hip/hip_runtime.h>

// QuantHotLowRank: out[N,1024] = quant(U)[ids] @ quant(B),  N=16384, R=64, D=1024
// f32 WMMA (V_WMMA_F32_16X16X4_F32) keeps exact f32 semantics of the reference.

typedef float v2f __attribute__((ext_vector_type(2)));
typedef float v4f __attribute__((ext_vector_type(4)));
typedef float v8f __attribute__((ext_vector_type(8)));

#define RR   64
#define DD   1024
#define QMAX 7.0f

// ---- group-quant helpers (group = 32 = one wave32) -------------------------

__device__ __forceinline__ float wave_amax32(float x) {
  float v = fabsf(x);
#pragma unroll
  for (int off = 16; off > 0; off >>= 1)
    v = fmaxf(v, __shfl_xor(v, off, 32));
  return v;
}

__device__ __forceinline__ float gquant(float w, float amax) {
  amax = fmaxf(amax, 1e-8f);                 // jnp.clip(amax, 1e-8)
  float scale = amax * (1.0f / QMAX);        // amax / 7
  float q = rintf(w / scale);                // round-half-to-even == jnp.round
  q = fminf(fmaxf(q, -QMAX), QMAX);
  return q * scale;
}

// ---- kernel 1: quantize B into K-pair-interleaved layout -------------------
// Bpk (as float2)[p*DD + n] = { Bq[2p][n], Bq[2p+1][n] },  p = 0..31
__global__ __launch_bounds__(256) void quantB_kernel(const float* __restrict__ B,
                                                     float* __restrict__ Bpk) {
  const int w    = (blockIdx.x * 256 + threadIdx.x) >> 5;   // 0..2047
  const int lane = threadIdx.x & 31;
  const int r    = w >> 5;                                  // row 0..63
  const int g    = w & 31;                                  // group 0..31
  const int col  = g * 32 + lane;
  const float x  = B[r * DD + col];
  const float q  = gquant(x, wave_amax32(x));
  Bpk[((((r >> 1) * DD) + col) << 1) + (r & 1)] = q;
}

// ---- kernel 2: gather + quantize selected U rows ---------------------------
__global__ __launch_bounds__(256) void quantU_kernel(const float* __restrict__ U,
                                                     const int* __restrict__ ids,
                                                     float* __restrict__ Uq, int N) {
  const int w    = (blockIdx.x * 256 + threadIdx.x) >> 5;   // wave id
  const int lane = threadIdx.x & 31;
  const int row  = w >> 1;                                  // gathered row
  if (row >= N) return;
  const int g    = w & 1;                                   // group 0/1 (R=64)
  const int col  = g * 32 + lane;
  const long long id = ids[row];
  const float x  = U[id * (long long)RR + col];
  const float q  = gquant(x, wave_amax32(x));
  Uq[row * RR + col] = q;
}

// ---- kernel 3: GEMM [N,64] x [64,1024] with V_WMMA_F32_16X16X4_F32 ---------
// 8 waves/block share one M-tile (A panel staged in LDS); each wave produces
// a 16x64 output strip (4 accumulators, 16 K-steps -> 64 WMMA ops).
__global__ __launch_bounds__(256) void gemm_kernel(const float* __restrict__ Uq,
                                                   const float* __restrict__ Bpk,
                                                   float* __restrict__ out) {
  __shared__ __attribute__((aligned(16))) float lA[16 * RR];   // 4 KB A panel

  const int tid   = threadIdx.x;
  const int lane  = tid & 31;
  const int wid   = blockIdx.x * 8 + (tid >> 5);
  const int nsup  = wid & 15;        // 64-col supertile: 0..15
  const int mtile = wid >> 4;        // 16-row tile (uniform across the block)

  // stage A panel: 1024 floats, 256 threads x float4, fully coalesced
  ((v4f*)lA)[tid] = ((const v4f*)(Uq + (size_t)mtile * 16 * RR))[tid];
  __syncthreads();

  const int mlane = lane & 15;       // M (lanes 0-15) per ISA A layout
  const int khalf = lane >> 4;       // selects K{0,1} vs K{2,3} within a frag

  // A fragments (16x4 each): VGPR0=K+0, VGPR1=K+1, lane-half adds +2
  v2f a[16];
#pragma unroll
  for (int kk = 0; kk < 16; ++kk)
    a[kk] = *(const v2f*)(lA + mlane * RR + kk * 4 + khalf * 2);

  v8f acc[4] = {};
  const int ncolBase = nsup * 64 + mlane;
#pragma unroll
  for (int kk = 0; kk < 16; ++kk) {
#pragma unroll
    for (int t = 0; t < 4; ++t) {
      // B frag: pair (K=4kk+2*khalf, +1) at column ncolBase + 16t — one 8B load
      const v2f b = *(const v2f*)(Bpk + (((size_t)(2 * kk + khalf) * DD
                                          + ncolBase + t * 16) << 1));
      acc[t] = __builtin_amdgcn_wmma_f32_16x16x4_f32(
          /*neg_a=*/false, a[kk], /*neg_b=*/false, b,
          /*c_mod=*/(short)0, acc[t], /*reuse_a=*/false, /*reuse_b=*/false);
    }
  }

  // C/D layout: VGPR j -> row (khalf*8 + j), lanes 0-15 -> columns
#pragma unroll
  for (int t = 0; t < 4; ++t) {
    const int colBase = nsup * 64 + t * 16 + mlane;
#pragma unroll
    for (int j = 0; j < 8; ++j) {
      const int row = mtile * 16 + khalf * 8 + j;
      out[(size_t)row * DD + colBase] = acc[t][j];
    }
  }
}

// ---- host launcher ---------------------------------------------------------
extern "C" void kernel_launch(void* const* d_in, const int* in_sizes, int n_in,
                              void* d_out, int out_size, void* d_ws, size_t ws_size,
                              hipStream_t stream) {
  (void)n_in; (void)out_size; (void)ws_size;
  const float* U   = (const float*)d_in[0];   // [200000, 64]
  const float* B   = (const float*)d_in[1];   // [64, 1024]
  const int*   ids = (const int*)d_in[2];     // [N]
  float*       out = (float*)d_out;           // [N, 1024]
  const int N = in_sizes[2];                  // 16384

  float* Bpk = (float*)d_ws;                  // 64*1024 f32 (256 KB)
  float* Uq  = Bpk + RR * DD;                 // N*64 f32 (4 MB)

  // 64 rows * 32 groups = 2048 waves -> 256 blocks of 256
  quantB_kernel<<<256, 256, 0, stream>>>(B, Bpk);

  // N rows * 2 groups waves -> N/4 blocks of 256
  quantU_kernel<<<(N * 2 * 32 + 255) / 256, 256, 0, stream>>>(U, ids, Uq, N);

  // (N/16) M-tiles * 16 supertiles waves, 8 waves/block
  const int waves = (N / 16) * 16;
  gemm_kernel<<<waves / 8, 256, 0, stream>>>(Uq, Bpk, out);
}